// NonLinearRE_41558103556709
// MI455X (gfx1250) — compile-verified
//
#include <hip/hip_runtime.h>
#include <math.h>

// ---------------------------------------------------------------------------
// Problem constants (match reference)
// ---------------------------------------------------------------------------
#define SEQ_LEN   4096
#define EMB       768
#define NPAIR     4096
#define NCHUNK    (NPAIR * 2)
#define MAX_ENT   16
#define MAX_CTX   50
#define POS_EMB   25
#define LINE_EMB  25
#define HID       100
#define NB_CLS    5
#define IN_FEAT   1711          // (768+25)*2 + 25 + 100
#define KPAD      1712          // IN_FEAT padded to multiple of 4 for WMMA K
#define NPADH     112           // HID padded to 7 * 16 WMMA N-tiles
#define CTX_OFF   1611          // ctx_out column offset in feat
#define LINE_OFF  1586          // line_emb column offset in feat

typedef __attribute__((ext_vector_type(2))) float v2f;
typedef __attribute__((ext_vector_type(8))) float v8f;

// ---------------------------------------------------------------------------
// Kernel 0: prepack W1 (1711x100) -> W1b, zero-padded to KPAD x NPADH, with
// K-pairs interleaved so a lane's B operand (two K rows, one column) is a
// single aligned float2 load:
//   element (k, n) lives at W1b[ ((k>>1)*NPADH + n)*2 + (k&1) ]
// ---------------------------------------------------------------------------
__global__ void prepack_w1(const float* __restrict__ W1, float* __restrict__ W1b) {
    int idx = blockIdx.x * blockDim.x + threadIdx.x;
    if (idx >= KPAD * NPADH) return;
    int k = idx / NPADH;
    int n = idx - k * NPADH;
    float v = (k < IN_FEAT && n < HID) ? W1[k * HID + n] : 0.0f;
    W1b[((size_t)(k >> 1) * NPADH + n) * 2 + (k & 1)] = v;
}

// ---------------------------------------------------------------------------
// Kernel 1: entity features. One block per chunk (2 chunks per pair).
// Writes avg_emb (768) + pos_emb (25) into feat row; second chunk also writes
// line_emb (25) and zeroes the single K-pad column.
// ---------------------------------------------------------------------------
__global__ void entity_feat(const float* __restrict__ tkn,
                            const int*   __restrict__ xr,
                            const float* __restrict__ pos_table,
                            const float* __restrict__ line_table,
                            float* __restrict__ feat) {
    int c = blockIdx.x;            // chunk id
    int p = c >> 1;
    int j = c & 1;
    const int* r = xr + p * 8 + j * 4;
    int s    = r[0];
    int e    = r[1];
    int sent = r[2];
    int len  = e - s + 1;
    int n    = len < MAX_ENT ? len : MAX_ENT;
    float inv = 1.0f / (float)len;

    float* frow = feat + (size_t)p * KPAD + j * (EMB + POS_EMB);

    // averaged token embedding, 3 dims per thread
    for (int d = threadIdx.x; d < EMB; d += blockDim.x) {
        float acc = 0.0f;
        for (int i = 0; i < n; ++i) {
            int gp = s + i;
            gp = gp < 0 ? 0 : (gp > SEQ_LEN - 1 ? SEQ_LEN - 1 : gp);
            acc += tkn[(size_t)gp * EMB + d];
        }
        frow[d] = acc * inv;
    }
    // positional embedding
    if (threadIdx.x < POS_EMB)
        frow[EMB + threadIdx.x] = pos_table[sent * POS_EMB + threadIdx.x];

    if (j == 1) {
        // line embedding: step1 - step2 + 5
        if (threadIdx.x >= 32 && threadIdx.x < 32 + LINE_EMB) {
            int t = threadIdx.x - 32;
            int li = xr[p * 8 + 3] - xr[p * 8 + 7] + 5;
            feat[(size_t)p * KPAD + LINE_OFF + t] = line_table[li * LINE_EMB + t];
        }
        if (threadIdx.x == 60)
            feat[(size_t)p * KPAD + IN_FEAT] = 0.0f;   // K-pad column
    }
}

// ---------------------------------------------------------------------------
// Kernel 2: fused context attention. One 256-thread block (8 waves) per pair.
// scores -> softmax -> c = attn @ ctx (768) -> ctx_out = c @ Wv (100),
// written straight into feat[:, 1611:1711].
// Uses (attn @ ctx) @ Wv == attn @ (ctx @ Wv)  => 33x fewer MACs.
// ---------------------------------------------------------------------------
__global__ void ctx_attn(const float* __restrict__ tkn,
                         const int*   __restrict__ xr,
                         const float* __restrict__ q,
                         const float* __restrict__ Wv,
                         float* __restrict__ feat) {
    __shared__ float sc[MAX_CTX];
    __shared__ float at[MAX_CTX];
    __shared__ int   ci[MAX_CTX];
    __shared__ float cvec[EMB];

    int p = blockIdx.x;
    const int* r = xr + p * 8;
    int s1 = r[0], e1 = r[1], s2 = r[4], e2 = r[5];
    bool fwd = (e1 <= s2);
    int cs = fwd ? e1 : e2;
    int ce = fwd ? s2 : s1;
    int L  = ce - cs;
    if (L > MAX_CTX) L = MAX_CTX;

    int lane = threadIdx.x & 31;
    int wave = threadIdx.x >> 5;

    // ---- scores: one wave per context row, lanes reduce over EMB ----
    for (int s = wave; s < MAX_CTX; s += 8) {
        int idx = cs + s;
        idx = idx < 0 ? 0 : (idx > SEQ_LEN - 1 ? SEQ_LEN - 1 : idx);
        const float* row = tkn + (size_t)idx * EMB;
        float acc = 0.0f;
        for (int e = lane; e < EMB; e += 32)
            acc += row[e] * q[e];
        for (int off = 16; off > 0; off >>= 1)
            acc += __shfl_xor(acc, off, 32);
        if (lane == 0) {
            ci[s] = idx;
            sc[s] = (s < L) ? acc : -1.0e9f;
        }
    }
    __syncthreads();

    // ---- softmax (tiny: every thread redundantly scans 50 values) ----
    float m = -1.0e30f;
    for (int s = 0; s < MAX_CTX; ++s) m = fmaxf(m, sc[s]);
    if (threadIdx.x < MAX_CTX)
        at[threadIdx.x] = __expf(sc[threadIdx.x] - m);
    __syncthreads();
    float denom = 0.0f;
    for (int s = 0; s < MAX_CTX; ++s) denom += at[s];
    float rden = 1.0f / denom;

    // ---- c[e] = sum_s attn_s * ctx[s][e] (masked rows contribute 0) ----
    for (int e = threadIdx.x; e < EMB; e += blockDim.x) {
        float acc = 0.0f;
        for (int s = 0; s < L; ++s)
            acc += at[s] * tkn[(size_t)ci[s] * EMB + e];
        cvec[e] = acc * rden;
    }
    __syncthreads();

    // ---- ctx_out = c @ Wv, 100 outputs ----
    if (threadIdx.x < HID) {
        float acc = 0.0f;
        for (int e = 0; e < EMB; ++e)
            acc += cvec[e] * Wv[e * HID + threadIdx.x];
        feat[(size_t)p * KPAD + CTX_OFF + threadIdx.x] = acc;
    }
}

// ---------------------------------------------------------------------------
// Kernel 3: Y = tanh(feat @ W1) @ W2 via V_WMMA_F32_16X16X4_F32.
// One block (8 waves) per 16-row strip; wave w (<7) owns N-tile w and runs
// 428 f32 WMMAs over K=1712. A lane layout (16x4 f32): lanes 0-15 rows M,
// K={0,1}; lanes 16-31 rows M, K={2,3}. B mirrors this via the prepacked W1b.
// tanh results go to LDS, then threads finish the 100x5 GEMM.
// ---------------------------------------------------------------------------
__global__ void gemm_wmma(const float* __restrict__ feat,
                          const float* __restrict__ W1b,
                          const float* __restrict__ W2,
                          float* __restrict__ out) {
    __shared__ float Hs[16 * NPADH];

    int lane = threadIdx.x & 31;
    int wave = threadIdx.x >> 5;
    int row0 = blockIdx.x * 16;
    int lo   = lane & 15;        // row (A) / column (B) within tile
    int hi   = lane >> 4;        // K sub-pair selector

    if (wave < 7) {
        const float* Ap = feat + (size_t)(row0 + lo) * KPAD + 2 * hi;
        const float* Bp = W1b + ((size_t)hi * NPADH + wave * 16 + lo) * 2;

        v8f acc = {};
#pragma unroll 4
        for (int k = 0; k < KPAD; k += 4) {
            v2f a = *(const v2f*)(Ap + k);
            v2f b = *(const v2f*)(Bp + (size_t)k * NPADH);
            acc = __builtin_amdgcn_wmma_f32_16x16x4_f32(
                false, a, false, b, (short)0, acc, false, false);
        }
        // scatter tanh(H) tile into LDS: D element (m,n): vgpr i holds
        // m = i + 8*hi, n = lo
#pragma unroll
        for (int i = 0; i < 8; ++i) {
            int mrow = i + 8 * hi;
            Hs[mrow * NPADH + wave * 16 + lo] = tanhf(acc[i]);
        }
    }
    __syncthreads();

    // ---- 16x5 output tile: y = Hs(:,0:100) @ W2(100x5) ----
    int t = threadIdx.x;
    if (t < 16 * NB_CLS) {
        int rrow = t / NB_CLS;
        int cls  = t - rrow * NB_CLS;
        float acc = 0.0f;
        for (int h = 0; h < HID; ++h)
            acc += Hs[rrow * NPADH + h] * W2[h * NB_CLS + cls];
        out[(size_t)(row0 + rrow) * NB_CLS + cls] = acc;
    }
}

// ---------------------------------------------------------------------------
// Kernel 4: pass-through copy of x_rels into the second output slot.
// ---------------------------------------------------------------------------
__global__ void copy_xrels(const int* __restrict__ xr, int* __restrict__ dst, int n) {
    int i = blockIdx.x * blockDim.x + threadIdx.x;
    if (i < n) dst[i] = xr[i];
}

// ---------------------------------------------------------------------------
// Launch
// ---------------------------------------------------------------------------
extern "C" void kernel_launch(void* const* d_in, const int* in_sizes, int n_in,
                              void* d_out, int out_size, void* d_ws, size_t ws_size,
                              hipStream_t stream) {
    const float* tkn        = (const float*)d_in[0];
    const int*   xr         = (const int*)  d_in[1];
    const float* pos_table  = (const float*)d_in[2];
    const float* line_table = (const float*)d_in[3];
    const float* q          = (const float*)d_in[4];
    const float* Wv         = (const float*)d_in[5];
    const float* W1         = (const float*)d_in[6];
    const float* W2         = (const float*)d_in[7];

    float* out = (float*)d_out;

    // workspace layout: feat (NPAIR x KPAD), then packed W1b (KPAD x NPADH)
    float* feat = (float*)d_ws;
    float* W1b  = feat + (size_t)NPAIR * KPAD;

    // 0) prepack W1
    {
        int n = KPAD * NPADH;
        prepack_w1<<<(n + 255) / 256, 256, 0, stream>>>(W1, W1b);
    }
    // 1) entity features -> feat[:, 0:1586] (+line, +pad col)
    entity_feat<<<NCHUNK, 256, 0, stream>>>(tkn, xr, pos_table, line_table, feat);
    // 2) fused attention -> feat[:, 1611:1711]
    ctx_attn<<<NPAIR, 256, 0, stream>>>(tkn, xr, q, Wv, feat);
    // 3) y_hat = tanh(feat @ W1) @ W2 via f32 WMMA
    gemm_wmma<<<NPAIR / 16, 256, 0, stream>>>(feat, W1b, W2, out);
    // 4) x_rels passthrough (second tuple output)
    {
        int n = NPAIR * 8;
        int* dst = (int*)(out + (size_t)NPAIR * NB_CLS);
        copy_xrels<<<(n + 255) / 256, 256, 0, stream>>>(xr, dst, n);
    }
}